// PeakbasedDetector_43542378447356
// MI455X (gfx1250) — compile-verified
//
#include <hip/hip_runtime.h>
#include <hip/hip_bf16.h>
#include <climits>

// ---------------------------------------------------------------------------
// PeakbasedDetector: per-row min/max -> normalize -> strict local maxima with
// 0.3 threshold -> mean inter-beat interval from first/last peak timestamps.
//
// MI455X strategy: HBM-bound streaming problem (147.5 MB of signal).
// - One workgroup per row, row staged into LDS ONCE via CDNA5 async
//   global->LDS copies (ASYNCcnt / s_wait_asynccnt path), then min/max,
//   normalize, and peak-scan all hit LDS only.
// - times[] is monotone: only 2 elements per row are ever read.
// ---------------------------------------------------------------------------

#ifndef __has_builtin
#define __has_builtin(x) 0
#endif

#if __has_builtin(__builtin_amdgcn_global_load_async_to_lds_b128) && \
    __has_builtin(__builtin_amdgcn_s_wait_asynccnt)
#define PEAK_ASYNC_LDS 1
#else
#define PEAK_ASYNC_LDS 0
#endif

#define PEAK_BS   512      // 16 wave32 per workgroup
#define PEAK_MAXN 9216     // LDS row capacity (floats): 36 KB

// Builtin signature (from clang diagnostic):
//   __builtin_amdgcn_global_load_async_to_lds_b128(int4 AS1*, int4 AS3*, Ii, Ii)
typedef int v4i_t __attribute__((__vector_size__(4 * sizeof(int))));
typedef __attribute__((address_space(1))) v4i_t gv4i_t;  // global int4
typedef __attribute__((address_space(3))) v4i_t lv4i_t;  // LDS int4

__global__ __launch_bounds__(PEAK_BS)
void peak_lds_kernel(const float* __restrict__ sig,
                     const float* __restrict__ times,
                     float* __restrict__ out, int N) {
    __shared__ __align__(16) float rowbuf[PEAK_MAXN];
    __shared__ float redf[2][PEAK_BS];
    __shared__ int   redi[3][PEAK_BS];

    const int tid = threadIdx.x;
    const long long base = (long long)blockIdx.x * (long long)N;
    const float* __restrict__ rowg = sig + base;

    // ---- stage row into LDS (single HBM read of the signal) ----
    const int nvec = N >> 2;  // 16-byte chunks
#if PEAK_ASYNC_LDS
    if (((((unsigned long long)(uintptr_t)rowg) & 15ull) == 0ull)) {
        for (int c = tid; c < nvec; c += PEAK_BS) {
            __builtin_amdgcn_global_load_async_to_lds_b128(
                (gv4i_t*)(rowg + 4 * c), (lv4i_t*)(rowbuf + 4 * c), 0, 0);
        }
        for (int i = (nvec << 2) + tid; i < N; i += PEAK_BS)
            rowbuf[i] = rowg[i];
        __builtin_amdgcn_s_wait_asynccnt(0);
    } else
#endif
    {
        for (int i = tid; i < N; i += PEAK_BS)
            rowbuf[i] = rowg[i];
    }
    __syncthreads();

    // ---- block min/max over the row ----
    float mn = 3.402823466e38f, mx = -3.402823466e38f;
    for (int i = tid; i < N; i += PEAK_BS) {
        float v = rowbuf[i];
        mn = fminf(mn, v);
        mx = fmaxf(mx, v);
    }
    redf[0][tid] = mn;
    redf[1][tid] = mx;
    __syncthreads();
    for (int s = PEAK_BS / 2; s > 0; s >>= 1) {
        if (tid < s) {
            redf[0][tid] = fminf(redf[0][tid], redf[0][tid + s]);
            redf[1][tid] = fmaxf(redf[1][tid], redf[1][tid + s]);
        }
        __syncthreads();
    }
    mn = redf[0][0];
    mx = redf[1][0];

    // ---- normalize in place, exactly as reference: (x - mn) / (mx - mn + eps)
    const float d = mx - mn + 1e-8f;
    for (int i = tid; i < N; i += PEAK_BS)
        rowbuf[i] = (rowbuf[i] - mn) / d;
    __syncthreads();

    // ---- strided peak scan: count, first index, last index ----
    int cnt = 0, fi = INT_MAX, li = -1;
    for (int i = tid; i < N; i += PEAK_BS) {
        if (i >= 1 && i <= N - 2) {
            float c = rowbuf[i];
            bool pk = (c > rowbuf[i - 1]) && (c >= rowbuf[i + 1]) && (c >= 0.3f);
            if (pk) {
                cnt++;
                fi = (i < fi) ? i : fi;
                li = (i > li) ? i : li;
            }
        }
    }
    redi[0][tid] = cnt;
    redi[1][tid] = fi;
    redi[2][tid] = li;
    __syncthreads();
    for (int s = PEAK_BS / 2; s > 0; s >>= 1) {
        if (tid < s) {
            redi[0][tid] += redi[0][tid + s];
            redi[1][tid] = min(redi[1][tid], redi[1][tid + s]);
            redi[2][tid] = max(redi[2][tid], redi[2][tid + s]);
        }
        __syncthreads();
    }

    if (tid == 0) {
        int count = redi[0][0];
        float r = 0.0f;
        if (count >= 2) {
            // times are monotone increasing: min/max over peaks == first/last peak
            float tf = times[base + redi[1][0]] * 1000.0f;
            float tl = times[base + redi[2][0]] * 1000.0f;
            r = (tl - tf) / (float)(count - 1);
        }
        out[blockIdx.x] = r;
    }
}

// Fallback for rows too large for LDS staging (not hit for N=9000).
__global__ __launch_bounds__(PEAK_BS)
void peak_global_kernel(const float* __restrict__ sig,
                        const float* __restrict__ times,
                        float* __restrict__ out, int N) {
    __shared__ float redf[2][PEAK_BS];
    __shared__ int   redi[3][PEAK_BS];

    const int tid = threadIdx.x;
    const long long base = (long long)blockIdx.x * (long long)N;
    const float* __restrict__ rowg = sig + base;

    float mn = 3.402823466e38f, mx = -3.402823466e38f;
    for (int i = tid; i < N; i += PEAK_BS) {
        float v = rowg[i];
        mn = fminf(mn, v);
        mx = fmaxf(mx, v);
    }
    redf[0][tid] = mn;
    redf[1][tid] = mx;
    __syncthreads();
    for (int s = PEAK_BS / 2; s > 0; s >>= 1) {
        if (tid < s) {
            redf[0][tid] = fminf(redf[0][tid], redf[0][tid + s]);
            redf[1][tid] = fmaxf(redf[1][tid], redf[1][tid + s]);
        }
        __syncthreads();
    }
    mn = redf[0][0];
    mx = redf[1][0];
    const float d = mx - mn + 1e-8f;

    int cnt = 0, fi = INT_MAX, li = -1;
    for (int i = tid; i < N; i += PEAK_BS) {
        if (i >= 1 && i <= N - 2) {
            float cm = (rowg[i - 1] - mn) / d;
            float cc = (rowg[i]     - mn) / d;
            float cp = (rowg[i + 1] - mn) / d;
            bool pk = (cc > cm) && (cc >= cp) && (cc >= 0.3f);
            if (pk) {
                cnt++;
                fi = (i < fi) ? i : fi;
                li = (i > li) ? i : li;
            }
        }
    }
    redi[0][tid] = cnt;
    redi[1][tid] = fi;
    redi[2][tid] = li;
    __syncthreads();
    for (int s = PEAK_BS / 2; s > 0; s >>= 1) {
        if (tid < s) {
            redi[0][tid] += redi[0][tid + s];
            redi[1][tid] = min(redi[1][tid], redi[1][tid + s]);
            redi[2][tid] = max(redi[2][tid], redi[2][tid + s]);
        }
        __syncthreads();
    }

    if (tid == 0) {
        int count = redi[0][0];
        float r = 0.0f;
        if (count >= 2) {
            float tf = times[base + redi[1][0]] * 1000.0f;
            float tl = times[base + redi[2][0]] * 1000.0f;
            r = (tl - tf) / (float)(count - 1);
        }
        out[blockIdx.x] = r;
    }
}

extern "C" void kernel_launch(void* const* d_in, const int* in_sizes, int n_in,
                              void* d_out, int out_size, void* d_ws, size_t ws_size,
                              hipStream_t stream) {
    const float* sig   = (const float*)d_in[0];
    const float* times = (const float*)d_in[1];
    float* out = (float*)d_out;

    const int B = out_size;              // 4096 rows
    if (B <= 0) return;
    const int N = in_sizes[0] / B;       // 9000 samples per row
    if (N <= 0) return;

    if (N <= PEAK_MAXN) {
        peak_lds_kernel<<<B, PEAK_BS, 0, stream>>>(sig, times, out, N);
    } else {
        peak_global_kernel<<<B, PEAK_BS, 0, stream>>>(sig, times, out, N);
    }
}